// ExpertRouter_41240275976829
// MI455X (gfx1250) — compile-verified
//
#include <hip/hip_runtime.h>
#include <math.h>

// Problem constants (fixed by the reference)
#define TOKENS   16384   // B*S = 4*4096
#define H_DIM    2048
#define E_EXP    8
#define TOPK     2
#define WAVES_PER_BLOCK 8          // 256 threads
#define TILE_TOKENS 16             // tokens per wave-group (WMMA M)
#define KSPLIT   4                 // waves cooperating on one tile (split-K)
#define KCOLS    (H_DIM / KSPLIT)  // 512 columns per wave
#define KBATCH   64                // columns staged per pipelined batch
#define NCHUNK   (KBATCH / 4)      // 16 WMMAs per batch
#define TILES_PER_BLOCK (WAVES_PER_BLOCK / KSPLIT)   // 2

typedef float v2f __attribute__((ext_vector_type(2)));
typedef float v8f __attribute__((ext_vector_type(8)));

__global__ void zero_counts_kernel(unsigned* __restrict__ counts) {
    if (threadIdx.x < E_EXP) counts[threadIdx.x] = 0u;
}

__global__ __launch_bounds__(256)
void router_wmma_kernel(const float* __restrict__ x,
                        const float* __restrict__ gate_w,
                        float* __restrict__ out_w,
                        int*   __restrict__ out_i,
                        unsigned* __restrict__ counts) {
    __shared__ float    s_red[WAVES_PER_BLOCK][32][8];   // 8 KB partial-C buffer
    __shared__ unsigned s_cnt[E_EXP];
    if (threadIdx.x < E_EXP) s_cnt[threadIdx.x] = 0u;

    const int lane   = threadIdx.x & 31;          // wave32
    const int wave   = threadIdx.x >> 5;
    const int kslice = wave & (KSPLIT - 1);       // which H-quarter this wave sums
    const int tile   = blockIdx.x * TILES_PER_BLOCK + (wave >> 2);
    const int half   = lane >> 4;                 // 0: K∈{0,1}, 1: K∈{2,3}
    const int e      = lane & 15;                 // expert column (valid < 8)

    // A row for this lane = token (tile*16 + (lane&15)); K offset = 2*half
    const float* __restrict__ xrow =
        x + (size_t)(tile * TILE_TOKENS + e) * H_DIM + kslice * KCOLS + 2 * half;
    // B: gate_w row (e mod 8) — lanes e>=8 duplicate experts; epilogue masks them
    const float* __restrict__ gwrow =
        gate_w + (size_t)(e & (E_EXP - 1)) * H_DIM + kslice * KCOLS + 2 * half;

    v8f c = {};  // 16x16 f32 partial accumulator (logits), 8 VGPRs

    for (int k0 = 0; k0 < KCOLS; k0 += KBATCH) {
        v2f xa[NCHUNK];
        v2f gb[NCHUNK];
        // burst of independent loads first -> many outstanding, deep MLP
#pragma unroll
        for (int u = 0; u < NCHUNK; ++u)
            xa[u] = __builtin_nontemporal_load(
                reinterpret_cast<const v2f*>(xrow + k0 + 4 * u));
#pragma unroll
        for (int u = 0; u < NCHUNK; ++u)
            gb[u] = *reinterpret_cast<const v2f*>(gwrow + k0 + 4 * u);
        // then the WMMA chain consumes them in issue order
#pragma unroll
        for (int u = 0; u < NCHUNK; ++u)
            c = __builtin_amdgcn_wmma_f32_16x16x4_f32(
                    false, xa[u], false, gb[u], (short)0, c, false, false);
    }

    // ---- split-K reduction through LDS ----
#pragma unroll
    for (int j = 0; j < 8; ++j) s_red[wave][lane][j] = c[j];
    __syncthreads();

    if (kslice == 0) {
        // sum the other three partials for this tile
#pragma unroll
        for (int w = 1; w < KSPLIT; ++w)
#pragma unroll
            for (int j = 0; j < 8; ++j) c[j] += s_red[wave + w][lane][j];

        // ---- epilogue: per C-row (token) top-2 over the 8 expert lanes ----
        const bool  evalid = (e < E_EXP);
        const float NEG = -__builtin_inff();
#pragma unroll
        for (int j = 0; j < 8; ++j) {
            float v1 = evalid ? c[j] : NEG;   // this lane's logit for expert e
            int   i1 = e;
            float v2 = NEG;
            int   i2 = E_EXP;                 // sentinel
#pragma unroll
            for (int m = 1; m <= 4; m <<= 1) {
                float ov1 = __shfl_xor(v1, m, 32);
                int   oi1 = __shfl_xor(i1, m, 32);
                float ov2 = __shfl_xor(v2, m, 32);
                int   oi2 = __shfl_xor(i2, m, 32);
                // merge two sorted top-2 pairs; ties -> lower index (stable)
                bool keepA = (v1 > ov1) || (v1 == ov1 && i1 < oi1);
                float nv1 = keepA ? v1 : ov1;  int ni1 = keepA ? i1 : oi1;
                float cv  = keepA ? ov1 : v1;  int ci  = keepA ? oi1 : i1;
                float sv  = keepA ? v2 : ov2;  int si  = keepA ? i2 : oi2;
                bool keepS = (sv > cv) || (sv == cv && si < ci);
                v1 = nv1; i1 = ni1;
                v2 = keepS ? sv : cv;
                i2 = keepS ? si : ci;
            }
            if ((lane & 15) == 0) {
                // token index: C-row M = j + 8*half within this tile
                const int t = tile * TILE_TOKENS + j + 8 * half;
                // renormalized top-2 softmax: denominator cancels
                const float d  = v2 - v1;          // <= 0
                const float ex = __expf(d);
                const float w2 = ex / (1.0f + ex);
                const float w1 = 1.0f - w2;
                out_w[2 * t + 0] = w1;
                out_w[2 * t + 1] = w2;
                out_i[2 * t + 0] = i1;
                out_i[2 * t + 1] = i2;
                atomicAdd(&s_cnt[i1], 1u);
                atomicAdd(&s_cnt[i2], 1u);
            }
        }
    }

    __syncthreads();
    if (threadIdx.x < E_EXP)
        atomicAdd(&counts[threadIdx.x], s_cnt[threadIdx.x]);
}

__global__ void aux_loss_kernel(const unsigned* __restrict__ counts,
                                float* __restrict__ aux_out) {
    float m[E_EXP];
    float mu = 0.0f;
#pragma unroll
    for (int e = 0; e < E_EXP; ++e) {
        m[e] = (float)counts[e] / (float)TOKENS;
        mu += m[e];
    }
    mu *= (1.0f / E_EXP);
    float var = 0.0f;
#pragma unroll
    for (int e = 0; e < E_EXP; ++e) {
        float d = m[e] - mu;
        var += d * d;
    }
    var *= (1.0f / (E_EXP - 1));       // ddof=1 (unbiased), as torch.var
    *aux_out = var * (float)E_EXP;
}

extern "C" void kernel_launch(void* const* d_in, const int* in_sizes, int n_in,
                              void* d_out, int out_size, void* d_ws, size_t ws_size,
                              hipStream_t stream) {
    (void)in_sizes; (void)n_in; (void)out_size; (void)ws_size;
    const float* x      = (const float*)d_in[0];   // (B,S,H) f32
    const float* gate_w = (const float*)d_in[1];   // (E,H)   f32

    // d_out flat layout (return order): weights (T*K f32) | indices (T*K i32) | aux (1 f32)
    float* out_w = (float*)d_out;
    int*   out_i = (int*)((float*)d_out + TOKENS * TOPK);
    float* aux   = (float*)d_out + 2 * TOKENS * TOPK;

    unsigned* counts = (unsigned*)d_ws;            // 8 x u32 scratch

    zero_counts_kernel<<<1, 32, 0, stream>>>(counts);

    const int tiles  = TOKENS / TILE_TOKENS;          // 1024 token tiles
    const int blocks = tiles / TILES_PER_BLOCK;       // 512 blocks x 256 threads
    router_wmma_kernel<<<blocks, 256, 0, stream>>>(x, gate_w, out_w, out_i, counts);

    aux_loss_kernel<<<1, 1, 0, stream>>>(counts, aux);
}